// RGCN_50818053046719
// MI455X (gfx1250) — compile-verified
//
#include <hip/hip_runtime.h>

#define NNODES 40000
#define NEDGES 640000
#define FDIM   128
#define NREL   4
#define LDSPITCH 132   // padded col-major pitch: conflict-free b64 B-fragment loads

typedef float v2f __attribute__((ext_vector_type(2)));
typedef float v8f __attribute__((ext_vector_type(8)));

// ---------------------------------------------------------------------------
// GEMM: Y[n][o] = sum_f X[n][f] * W[f][o]   (128x128 W in padded col-major LDS)
// grid = (ceil(NNODES/128), NREL+1); block = 128 threads = 4 waves.
// Each wave computes TWO 16-row x 128-col strips (32 rows), sharing every
// B fragment across both strips: 16 v_wmma per 8 LDS b64 loads.
// blockIdx.y < NREL  -> W = Wrel[r],  Y = Yrel + r*N*128
// blockIdx.y == NREL -> W = Wroot,    Y = Yroot (initializes agg)
// ---------------------------------------------------------------------------
__global__ __launch_bounds__(128) void rgcn_gemm_wmma(
    const float* __restrict__ X, const float* __restrict__ Wrel,
    const float* __restrict__ Wroot, float* __restrict__ Yrel,
    float* __restrict__ Yroot)
{
    __shared__ float sW[FDIM * LDSPITCH];

    const int tid   = threadIdx.x;
    const int wave  = tid >> 5;
    const int lane  = tid & 31;
    const int col   = lane & 15;   // N-index in 16-wide tile / M-row for A
    const int khalf = lane >> 4;   // lanes 16-31 carry the upper K pair

    const int r = blockIdx.y;
    const float* __restrict__ W = (r < NREL) ? (Wrel + (size_t)r * FDIM * FDIM)
                                             : Wroot;
    float* __restrict__ Y = (r < NREL) ? (Yrel + (size_t)r * NNODES * FDIM)
                                       : Yroot;

    // Stage W into LDS, transposing to padded column-major:
    // sW[col*LDSPITCH + k] = W[k][col]. Global reads stay 128-bit coalesced.
    {
        const float4* __restrict__ W4 = (const float4*)W;
#pragma unroll
        for (int it = 0; it < (FDIM * FDIM / 4) / 128; ++it) {
            const int i  = tid + it * 128;
            const int k  = i >> 5;          // W row (K index)
            const int c4 = (i & 31) * 4;    // starting W column
            const float4 w4 = W4[i];
            sW[(c4 + 0) * LDSPITCH + k] = w4.x;
            sW[(c4 + 1) * LDSPITCH + k] = w4.y;
            sW[(c4 + 2) * LDSPITCH + k] = w4.z;
            sW[(c4 + 3) * LDSPITCH + k] = w4.w;
        }
    }
    __syncthreads();

    const int row0 = (blockIdx.x * 4 + wave) * 32;

    // A-fragment rows for the two strips (clamped; stores are guarded).
    int ra0 = row0 + col;       if (ra0 >= NNODES) ra0 = NNODES - 1;
    int ra1 = row0 + 16 + col;  if (ra1 >= NNODES) ra1 = NNODES - 1;
    const float* __restrict__ X0 = X + (size_t)ra0 * FDIM + 2 * khalf;
    const float* __restrict__ X1 = X + (size_t)ra1 * FDIM + 2 * khalf;

    v8f acc0[8] = {};
    v8f acc1[8] = {};

    for (int k0 = 0; k0 < FDIM; k0 += 4) {
        const int kb = k0 + 2 * khalf;
        // A 16x4 f32 fragments: lanes 0-15 K={k0,k0+1}, lanes 16-31 K={k0+2,k0+3}
        const float2 a0v = *(const float2*)(X0 + k0);
        const float2 a1v = *(const float2*)(X1 + k0);
        v2f a0; a0.x = a0v.x; a0.y = a0v.y;
        v2f a1; a1.x = a1v.x; a1.y = a1v.y;
#pragma unroll
        for (int t = 0; t < 8; ++t) {
            // B 4x16 fragment, column tile t: one aligned b64 LDS load.
            const float2 bv =
                *(const float2*)(sW + (t * 16 + col) * LDSPITCH + kb);
            v2f b; b.x = bv.x; b.y = bv.y;
            acc0[t] = __builtin_amdgcn_wmma_f32_16x16x4_f32(
                false, a0, false, b, (short)0, acc0[t], false, false);
            acc1[t] = __builtin_amdgcn_wmma_f32_16x16x4_f32(
                false, a1, false, b, (short)0, acc1[t], false, false);
        }
    }

    // C/D layout: VGPR v holds row (v + 8*khalf), N = col.
    if (row0 + 32 <= NNODES) {
#pragma unroll
        for (int t = 0; t < 8; ++t) {
#pragma unroll
            for (int v = 0; v < 8; ++v) {
                const size_t rr = (size_t)(row0 + v + 8 * khalf);
                Y[rr * FDIM + t * 16 + col]              = acc0[t][v];
                Y[(rr + 16) * FDIM + t * 16 + col]       = acc1[t][v];
            }
        }
    } else {
#pragma unroll
        for (int t = 0; t < 8; ++t) {
#pragma unroll
            for (int v = 0; v < 8; ++v) {
                const int r0s = row0 + v + 8 * khalf;
                const int r1s = r0s + 16;
                if (r0s < NNODES)
                    Y[(size_t)r0s * FDIM + t * 16 + col] = acc0[t][v];
                if (r1s < NNODES)
                    Y[(size_t)r1s * FDIM + t * 16 + col] = acc1[t][v];
            }
        }
    }
}

// ---------------------------------------------------------------------------
// Per-(dst, relation) edge counting and reciprocal (shared by both layers).
// ---------------------------------------------------------------------------
__global__ void zero_u32(unsigned* __restrict__ p, int n)
{
    int i = blockIdx.x * blockDim.x + threadIdx.x;
    if (i < n) p[i] = 0u;
}

__global__ void count_edges(const int* __restrict__ ei,
                            const int* __restrict__ et,
                            unsigned* __restrict__ cnt)
{
    int e = blockIdx.x * blockDim.x + threadIdx.x;
    if (e < NEDGES) {
        int d = ei[NEDGES + e];
        int r = et[e];
        atomicAdd(&cnt[d * NREL + r], 1u);
    }
}

__global__ void recip_counts(const unsigned* __restrict__ cnt,
                             float* __restrict__ inv, int n)
{
    int i = blockIdx.x * blockDim.x + threadIdx.x;
    if (i < n) inv[i] = 1.0f / fmaxf((float)cnt[i], 1.0f);
}

// ---------------------------------------------------------------------------
// Scatter: one edge per wave; 128-bit gather of the pre-transformed source
// feature, scale by 1/count(dst,rel), no-return float atomic-add into agg.
// block = 256 threads = 8 edges.
// ---------------------------------------------------------------------------
__global__ __launch_bounds__(256) void rgcn_scatter(
    const float* __restrict__ hrel, const int* __restrict__ ei,
    const int* __restrict__ et, const float* __restrict__ inv_cnt,
    float* __restrict__ agg)
{
    const int wave = threadIdx.x >> 5;
    const int lane = threadIdx.x & 31;
    const int e = blockIdx.x * 8 + wave;
    if (e >= NEDGES) return;

    const int s = ei[e];
    const int d = ei[NEDGES + e];
    const int r = et[e];
    const float scale = inv_cnt[d * NREL + r];

    const float4 v =
        ((const float4*)(hrel + ((size_t)r * NNODES + (size_t)s) * FDIM))[lane];
    float* dsta = agg + (size_t)d * FDIM + lane * 4;
    unsafeAtomicAdd(dsta + 0, v.x * scale);
    unsafeAtomicAdd(dsta + 1, v.y * scale);
    unsafeAtomicAdd(dsta + 2, v.z * scale);
    unsafeAtomicAdd(dsta + 3, v.w * scale);
}

// ---------------------------------------------------------------------------
// out = (agg + bias)  [+ ReLU for layer 1]
// ---------------------------------------------------------------------------
__global__ void bias_act(const float* __restrict__ a,
                         const float* __restrict__ bias,
                         float* __restrict__ o, int relu)
{
    int i = blockIdx.x * blockDim.x + threadIdx.x;
    float v = a[i] + bias[i & (FDIM - 1)];
    o[i] = relu ? fmaxf(v, 0.0f) : v;
}

// ---------------------------------------------------------------------------
extern "C" void kernel_launch(void* const* d_in, const int* in_sizes, int n_in,
                              void* d_out, int out_size, void* d_ws, size_t ws_size,
                              hipStream_t stream)
{
    const float* x     = (const float*)d_in[0];
    const float* w1    = (const float*)d_in[1];
    const float* root1 = (const float*)d_in[2];
    const float* b1    = (const float*)d_in[3];
    const float* w2    = (const float*)d_in[4];
    const float* root2 = (const float*)d_in[5];
    const float* b2    = (const float*)d_in[6];
    const int*   ei    = (const int*)d_in[7];   // [2, E]
    const int*   et    = (const int*)d_in[8];   // [E]
    float*       out   = (float*)d_out;

    // Workspace layout (floats): h_rel[4*N*128] | agg[N*128] | h[N*128] |
    //                            inv[N*R] | cnt[N*R]
    float* ws     = (float*)d_ws;
    float* h_rel  = ws;
    float* agg    = h_rel + (size_t)NREL * NNODES * FDIM;
    float* h      = agg + (size_t)NNODES * FDIM;
    float* inv    = h + (size_t)NNODES * FDIM;
    unsigned* cnt = (unsigned*)(inv + (size_t)NNODES * NREL);

    const int NR = NNODES * NREL;

    // Edge counts (layer-invariant).
    zero_u32<<<(NR + 255) / 256, 256, 0, stream>>>(cnt, NR);
    count_edges<<<(NEDGES + 255) / 256, 256, 0, stream>>>(ei, et, cnt);
    recip_counts<<<(NR + 255) / 256, 256, 0, stream>>>(cnt, inv, NR);

    const dim3 ggrid((NNODES + 127) / 128, NREL + 1);

    // ---- Layer 1 ----
    rgcn_gemm_wmma<<<ggrid, 128, 0, stream>>>(x, w1, root1, h_rel, agg);
    rgcn_scatter<<<NEDGES / 8, 256, 0, stream>>>(h_rel, ei, et, inv, agg);
    bias_act<<<(NNODES * FDIM) / 256, 256, 0, stream>>>(agg, b1, h, 1);

    // ---- Layer 2 ----
    rgcn_gemm_wmma<<<ggrid, 128, 0, stream>>>(h, w2, root2, h_rel, agg);
    rgcn_scatter<<<NEDGES / 8, 256, 0, stream>>>(h_rel, ei, et, inv, agg);
    bias_act<<<(NNODES * FDIM) / 256, 256, 0, stream>>>(agg, b2, out, 0);
}